// D4A_15169824489702
// MI455X (gfx1250) — compile-verified
//
#include <hip/hip_runtime.h>

typedef __attribute__((ext_vector_type(2))) float v2f;
typedef __attribute__((ext_vector_type(8))) float v8f;

#define IN   128
#define HID  64
#define OUTF 40

// Native CDNA5 float atomic-max (no return -> STOREcnt; s_endpgm waits idle).
__device__ __forceinline__ void atomic_max_f32(float* p, float v) {
  asm volatile("global_atomic_max_num_f32 %0, %1, off scope:SCOPE_DEV"
               :: "v"(p), "v"(v) : "memory");
}

// ---------------------------------------------------------------- init
__global__ void D4A_init_k(float* __restrict__ aggs, long nAgg,
                           int* __restrict__ cnt, int nCnt) {
  long i = (long)blockIdx.x * blockDim.x + threadIdx.x;
  long stride = (long)gridDim.x * blockDim.x;
  for (long j = i; j < nAgg; j += stride) aggs[j] = -__builtin_inff();
  for (long j = i; j < nCnt; j += stride) cnt[j] = 0;
}

// ------------------------------------------------- edge scatter, layer 1
// One wave per edge: 128 features, 4 per lane (float4 coalesced read).
__global__ __launch_bounds__(256)
void D4A_scatter1_k(const float* __restrict__ x, const int* __restrict__ src,
                    const int* __restrict__ dst, float* __restrict__ agg,
                    int* __restrict__ cnt, int nE) {
  int w = (blockIdx.x * 256 + threadIdx.x) >> 5;
  int lane = threadIdx.x & 31;
  if (w >= nE) return;
  int s = src[w], d = dst[w];
  const float4 xv = *(const float4*)(x + (long)s * IN + lane * 4);
  float* ap = agg + (long)d * IN + lane * 4;
  atomic_max_f32(ap + 0, xv.x);
  atomic_max_f32(ap + 1, xv.y);
  atomic_max_f32(ap + 2, xv.z);
  atomic_max_f32(ap + 3, xv.w);
  if (lane == 0) atomicAdd(cnt + d, 1);
}

// ------------------------------------------------- edge scatter, layer 2
__global__ __launch_bounds__(256)
void D4A_scatter2_k(const float* __restrict__ h1, const int* __restrict__ src,
                    const int* __restrict__ dst, float* __restrict__ agg, int nE) {
  int w = (blockIdx.x * 256 + threadIdx.x) >> 5;
  int lane = threadIdx.x & 31;
  if (w >= nE) return;
  int s = src[w], d = dst[w];
  const v2f hv = *(const v2f*)(h1 + (long)s * HID + lane * 2);
  float* ap = agg + (long)d * HID + lane * 2;
  atomic_max_f32(ap + 0, hv.x);
  atomic_max_f32(ap + 1, hv.y);
}

// --------------------------------- layer 1: conv + GEMM(128->64) + L2 + ReLU
// One wave = one 16-node tile. FP32 WMMA 16x16x4, K swept in 32 steps.
__global__ __launch_bounds__(256)
void D4A_gemm1_k(const float* __restrict__ x, const float* __restrict__ agg,
                 const int* __restrict__ cnt, const float* __restrict__ W1,
                 float* __restrict__ h1, int N, int ntiles) {
  __shared__ v2f wlds[32 * 4 * 32];   // [kk][nt][lane] packed B fragments, 32KB
  int tid = threadIdx.x;
  for (int e = tid; e < 32 * 4 * 32; e += 256) {
    int kk = e >> 7, rem = e & 127, nt = rem >> 5, l = rem & 31;
    int k = 4 * kk + ((l & 16) ? 2 : 0);
    int n = nt * 16 + (l & 15);
    v2f v;
    v.x = W1[k * HID + n];
    v.y = W1[(k + 1) * HID + n];
    wlds[e] = v;
  }
  __syncthreads();

  int wave = tid >> 5, lane = tid & 31;
  int tile = blockIdx.x * 8 + wave;
  if (tile >= ntiles) return;                 // wave-uniform guard
  long base = (long)tile * 16;
  long rowA = base + (lane & 15);
  if (rowA > (long)N - 1) rowA = (long)N - 1; // clamp (N%16==0 normally)
  int koff = (lane & 16) ? 2 : 0;
  const float* xr = x + rowA * IN + koff;
  const float* ar = agg + rowA * IN + koff;
  bool cpos = cnt[rowA] > 0;

  v8f acc0 = {}, acc1 = {}, acc2 = {}, acc3 = {};
  for (int kk = 0; kk < 32; ++kk) {
    v2f xv = *(const v2f*)(xr + 4 * kk);
    v2f av = *(const v2f*)(ar + 4 * kk);
    v2f a;
    a.x = cpos ? (2.0f * xv.x - av.x) : xv.x;
    a.y = cpos ? (2.0f * xv.y - av.y) : xv.y;
    const v2f* wb = &wlds[kk * 128 + lane];
    acc0 = __builtin_amdgcn_wmma_f32_16x16x4_f32(false, a, false, wb[0],  (short)0, acc0, false, false);
    acc1 = __builtin_amdgcn_wmma_f32_16x16x4_f32(false, a, false, wb[32], (short)0, acc1, false, false);
    acc2 = __builtin_amdgcn_wmma_f32_16x16x4_f32(false, a, false, wb[64], (short)0, acc2, false, false);
    acc3 = __builtin_amdgcn_wmma_f32_16x16x4_f32(false, a, false, wb[96], (short)0, acc3, false, false);
  }

  // Row-wise L2 normalize (64 cols) + ReLU. Row r lives in element r of each
  // accumulator; lanes 0-15 hold rows 0-7, lanes 16-31 hold rows 8-15.
  float invn[8];
#pragma unroll
  for (int r = 0; r < 8; ++r) {
    float ss = acc0[r] * acc0[r] + acc1[r] * acc1[r] +
               acc2[r] * acc2[r] + acc3[r] * acc3[r];
    ss += __shfl_xor(ss, 1, 32);
    ss += __shfl_xor(ss, 2, 32);
    ss += __shfl_xor(ss, 4, 32);
    ss += __shfl_xor(ss, 8, 32);
    invn[r] = 1.0f / fmaxf(sqrtf(ss), 1e-12f);
  }
  int lcol = lane & 15;
#pragma unroll
  for (int r = 0; r < 8; ++r) {
    int m = (lane < 16) ? r : (8 + r);
    long node = base + m;
    if (node < N) {
      float* hp = h1 + node * HID + lcol;
      hp[0]  = fmaxf(acc0[r] * invn[r], 0.0f);
      hp[16] = fmaxf(acc1[r] * invn[r], 0.0f);
      hp[32] = fmaxf(acc2[r] * invn[r], 0.0f);
      hp[48] = fmaxf(acc3[r] * invn[r], 0.0f);
    }
  }
}

// --------------------------------- layer 2: conv + GEMM(64->40) -> out
__global__ __launch_bounds__(256)
void D4A_gemm2_k(const float* __restrict__ h1, const float* __restrict__ agg,
                 const int* __restrict__ cnt, const float* __restrict__ W2,
                 float* __restrict__ out, int N, int ntiles) {
  __shared__ v2f wlds[16 * 3 * 32];   // 12KB; cols >= 40 zero-padded
  int tid = threadIdx.x;
  for (int e = tid; e < 16 * 3 * 32; e += 256) {
    int kk = e / 96, rem = e % 96, nt = rem >> 5, l = rem & 31;
    int k = 4 * kk + ((l & 16) ? 2 : 0);
    int n = nt * 16 + (l & 15);
    v2f v;
    v.x = (n < OUTF) ? W2[k * OUTF + n] : 0.0f;
    v.y = (n < OUTF) ? W2[(k + 1) * OUTF + n] : 0.0f;
    wlds[e] = v;
  }
  __syncthreads();

  int wave = tid >> 5, lane = tid & 31;
  int tile = blockIdx.x * 8 + wave;
  if (tile >= ntiles) return;
  long base = (long)tile * 16;
  long rowA = base + (lane & 15);
  if (rowA > (long)N - 1) rowA = (long)N - 1;
  int koff = (lane & 16) ? 2 : 0;
  const float* hr = h1 + rowA * HID + koff;
  const float* ar = agg + rowA * HID + koff;
  bool cpos = cnt[rowA] > 0;

  v8f acc0 = {}, acc1 = {}, acc2 = {};
  for (int kk = 0; kk < 16; ++kk) {
    v2f hv = *(const v2f*)(hr + 4 * kk);
    v2f av = *(const v2f*)(ar + 4 * kk);
    v2f a;
    a.x = cpos ? (2.0f * hv.x - av.x) : hv.x;
    a.y = cpos ? (2.0f * hv.y - av.y) : hv.y;
    const v2f* wb = &wlds[kk * 96 + lane];
    acc0 = __builtin_amdgcn_wmma_f32_16x16x4_f32(false, a, false, wb[0],  (short)0, acc0, false, false);
    acc1 = __builtin_amdgcn_wmma_f32_16x16x4_f32(false, a, false, wb[32], (short)0, acc1, false, false);
    acc2 = __builtin_amdgcn_wmma_f32_16x16x4_f32(false, a, false, wb[64], (short)0, acc2, false, false);
  }

  int lcol = lane & 15;
#pragma unroll
  for (int r = 0; r < 8; ++r) {
    int m = (lane < 16) ? r : (8 + r);
    long node = base + m;
    if (node < N) {
      float* op = out + node * OUTF + lcol;
      op[0]  = acc0[r];
      op[16] = acc1[r];
      if (lcol < 8) op[32] = acc2[r];
    }
  }
}

// ---------------------------------------------------------------- launch
extern "C" void kernel_launch(void* const* d_in, const int* in_sizes, int n_in,
                              void* d_out, int out_size, void* d_ws, size_t ws_size,
                              hipStream_t stream) {
  const float* x  = (const float*)d_in[0];
  const int*   ei = (const int*)d_in[1];
  const float* W1 = (const float*)d_in[2];
  const float* W2 = (const float*)d_in[3];
  float* out = (float*)d_out;

  int N = in_sizes[0] / IN;
  int E = in_sizes[1] / 2;
  const int* src = ei;
  const int* dst = ei + E;

  // workspace layout: agg1[N*128] | agg2[N*64] | h1[N*64] | cnt[N]
  float* agg1 = (float*)d_ws;
  float* agg2 = agg1 + (size_t)N * IN;
  float* h1   = agg2 + (size_t)N * HID;
  int*   cnt  = (int*)(h1 + (size_t)N * HID);

  long nAgg = (long)N * (IN + HID);           // agg1+agg2 are contiguous
  D4A_init_k<<<2048, 256, 0, stream>>>(agg1, nAgg, cnt, N);

  int edgeBlocks = (E + 7) / 8;               // 8 edge-waves per 256-thr block
  D4A_scatter1_k<<<edgeBlocks, 256, 0, stream>>>(x, src, dst, agg1, cnt, E);

  int ntiles = (N + 15) / 16;
  int gemmBlocks = (ntiles + 7) / 8;
  D4A_gemm1_k<<<gemmBlocks, 256, 0, stream>>>(x, agg1, cnt, W1, h1, N, ntiles);

  D4A_scatter2_k<<<edgeBlocks, 256, 0, stream>>>(h1, src, dst, agg2, E);

  D4A_gemm2_k<<<gemmBlocks, 256, 0, stream>>>(h1, agg2, cnt, W2, out, N, ntiles);
}